// SSGC_25915832664730
// MI455X (gfx1250) — compile-verified
//
#include <hip/hip_runtime.h>
#include <hip/hip_bf16.h>

typedef float v2f __attribute__((ext_vector_type(2)));
typedef float v8f __attribute__((ext_vector_type(8)));

#define DFEAT 64
#define NCLS  40
#define KSTEP 16
#define ALPHA_C 0.05f

// ---------------------------------------------------------------------------
// init: deg = 1.0 (self loop), cnt = 0, h = 0
// ---------------------------------------------------------------------------
__global__ void ssgc_init(float* __restrict__ deg, int* __restrict__ cnt,
                          float* __restrict__ h, int n, long long nh) {
  long long t = (long long)blockIdx.x * blockDim.x + threadIdx.x;
  if (t < n) { deg[t] = 1.0f; cnt[t] = 0; }
  if (t < nh) h[t] = 0.0f;
}

// ---------------------------------------------------------------------------
// degree + per-dst edge counts
// ---------------------------------------------------------------------------
__global__ void ssgc_count(const int* __restrict__ dst, float* __restrict__ deg,
                           int* __restrict__ cnt, int e) {
  int t = blockIdx.x * blockDim.x + threadIdx.x;
  if (t < e) {
    int d = dst[t];
    atomicAdd(&deg[d], 1.0f);
    atomicAdd(&cnt[d], 1);
  }
}

// dinv = rsqrt(deg) in place
__global__ void ssgc_dinv(float* __restrict__ deg, int n) {
  int t = blockIdx.x * blockDim.x + threadIdx.x;
  if (t < n) deg[t] = rsqrtf(deg[t]);
}

// ---------------------------------------------------------------------------
// single-workgroup chunked exclusive scan of cnt -> row_ptr (and cursor copy)
// 98 chunks of 1024; cost trivial vs 16 propagation rounds
// ---------------------------------------------------------------------------
__global__ void ssgc_scan(const int* __restrict__ cnt, int* __restrict__ rowp,
                          int* __restrict__ cursor, int n, int e) {
  __shared__ int sh[1024];
  __shared__ int carry;
  if (threadIdx.x == 0) carry = 0;
  __syncthreads();
  for (int base = 0; base < n; base += 1024) {
    int i = base + (int)threadIdx.x;
    int v = (i < n) ? cnt[i] : 0;
    sh[threadIdx.x] = v;
    __syncthreads();
    for (int off = 1; off < 1024; off <<= 1) {
      int t = (threadIdx.x >= (unsigned)off) ? sh[threadIdx.x - off] : 0;
      __syncthreads();
      sh[threadIdx.x] += t;
      __syncthreads();
    }
    int excl = sh[threadIdx.x] - v + carry;
    if (i < n) { rowp[i] = excl; cursor[i] = excl; }
    __syncthreads();
    if (threadIdx.x == 0) carry += sh[1023];
    __syncthreads();
  }
  if (threadIdx.x == 0) rowp[n] = e;
}

// ---------------------------------------------------------------------------
// scatter edges into CSR slots; precompute normalized edge weights
// ---------------------------------------------------------------------------
__global__ void ssgc_scatter(const int* __restrict__ src, const int* __restrict__ dst,
                             const float* __restrict__ dinv, int* __restrict__ cursor,
                             int* __restrict__ col, float* __restrict__ wgt, int e) {
  int t = blockIdx.x * blockDim.x + threadIdx.x;
  if (t < e) {
    int d = dst[t], s = src[t];
    int pos = atomicAdd(&cursor[d], 1);
    col[pos] = s;
    wgt[pos] = dinv[d] * dinv[s];
  }
}

// ---------------------------------------------------------------------------
// one propagation round, fused with h update.
// one wave32 per node; each lane owns 2 consecutive floats of the 64-wide row
// -> every gather/store is a fully coalesced 256B row access (L2-resident).
// ---------------------------------------------------------------------------
__global__ void ssgc_step(const float* __restrict__ xin, float* __restrict__ xout,
                          float* __restrict__ h, const float* __restrict__ feat,
                          const float* __restrict__ dinv, const int* __restrict__ rowp,
                          const int* __restrict__ col, const float* __restrict__ wgt,
                          int n) {
  int i = (blockIdx.x * blockDim.x + threadIdx.x) >> 5;   // wave id == node id
  if (i >= n) return;
  int lane = threadIdx.x & 31;
  int off  = lane * 2;

  int beg = rowp[i];
  int end = rowp[i + 1];
  v2f acc = {0.0f, 0.0f};
  for (int ee = beg; ee < end; ++ee) {
    int   s  = col[ee];
    float wv = wgt[ee];
    if (ee + 1 < end) {
      // next source row lives in L2; pull it toward the WGP to hide latency
      __builtin_prefetch(xin + (size_t)col[ee + 1] * DFEAT, 0, 0);
    }
    v2f xv = *(const v2f*)(xin + (size_t)s * DFEAT + off);
    acc += wv * xv;
  }
  float di = dinv[i];
  v2f xself = *(const v2f*)(xin + (size_t)i * DFEAT + off);
  acc += (di * di) * xself;

  *(v2f*)(xout + (size_t)i * DFEAT + off) = acc;

  v2f fv = *(const v2f*)(feat + (size_t)i * DFEAT + off);
  v2f hv = *(const v2f*)(h + (size_t)i * DFEAT + off);
  hv = (hv + (1.0f - ALPHA_C) * acc + ALPHA_C * fv) * (1.0f / (float)KSTEP);
  *(v2f*)(h + (size_t)i * DFEAT + off) = hv;
}

// ---------------------------------------------------------------------------
// out[N,40] = h[N,64] @ W[40,64]^T + b    via V_WMMA_F32_16X16X4_F32
// one wave per 16x16 output tile; 16 k-steps of K=4 cover the 64-deep dot.
// A frag (16x4 f32): lane m=l%16, VGPR0/1 = K {0,1} (lanes 0-15) / {2,3} (16-31)
// B frag (4x16 f32): lane n=l%16, same K split.
// C/D  (16x16 f32): VGPR r -> M=r (lanes 0-15), M=r+8 (lanes 16-31), N=l%16.
// EXEC stays all-1s: tile-uniform early-out, branchless zero-mask for n>=40.
// ---------------------------------------------------------------------------
__global__ void ssgc_gemm(const float* __restrict__ h, const float* __restrict__ W,
                          const float* __restrict__ bias, float* __restrict__ out,
                          int n) {
  const int TN = 3;                       // ceil(40/16) column tiles
  int gw = (blockIdx.x * blockDim.x + threadIdx.x) >> 5;
  int tilesM = (n + 15) >> 4;
  int tm = gw / TN;
  int tn = gw - tm * TN;
  if (tm >= tilesM) return;               // wave-uniform

  int lane = threadIdx.x & 31;
  int half = lane >> 4;                   // K sub-pair selector
  int l16  = lane & 15;

  int m  = tm * 16 + l16;
  int mc = m < n ? m : n - 1;             // clamped load row (branchless)
  int nn = tn * 16 + l16;
  float nsel = (nn < NCLS) ? 1.0f : 0.0f;
  int   nc   = (nn < NCLS) ? nn : 0;

  const float* ha = h + (size_t)mc * DFEAT + half * 2;
  const float* wb = W + (size_t)nc * DFEAT + half * 2;

  v8f acc = {0.f, 0.f, 0.f, 0.f, 0.f, 0.f, 0.f, 0.f};
#pragma unroll
  for (int k = 0; k < DFEAT / 4; ++k) {
    v2f a = *(const v2f*)(ha + k * 4);
    v2f b = *(const v2f*)(wb + k * 4);
    b *= nsel;                            // zero padded columns, no divergence
    acc = __builtin_amdgcn_wmma_f32_16x16x4_f32(
        /*neg_a=*/false, a, /*neg_b=*/false, b,
        /*c_mod=*/(short)0, acc, /*reuse_a=*/false, /*reuse_b=*/false);
  }

  if (nn < NCLS) {
    float bb = bias[nn];
#pragma unroll
    for (int r = 0; r < 8; ++r) {
      int mm = tm * 16 + half * 8 + r;
      if (mm < n) out[(size_t)mm * NCLS + nn] = acc[r] + bb;
    }
  }
}

// ---------------------------------------------------------------------------
extern "C" void kernel_launch(void* const* d_in, const int* in_sizes, int n_in,
                              void* d_out, int out_size, void* d_ws, size_t ws_size,
                              hipStream_t stream) {
  (void)n_in; (void)out_size; (void)ws_size;
  const float* feat = (const float*)d_in[0];   // [N,64]
  const float* W    = (const float*)d_in[1];   // [40,64]
  const float* bias = (const float*)d_in[2];   // [40]
  const int*   src  = (const int*)d_in[3];     // [E]
  const int*   dst  = (const int*)d_in[4];     // [E]
  float* out = (float*)d_out;                  // [N,40]

  const int n = in_sizes[0] / DFEAT;
  const int e = in_sizes[3];

  // workspace carve-out (256B aligned slices)
  char* ws = (char*)d_ws;
  size_t p = 0;
  auto carve = [&](size_t bytes) -> void* {
    void* r = ws + p;
    p = (p + bytes + 255) & ~(size_t)255;
    return r;
  };
  float* dinv   = (float*)carve((size_t)n * 4);
  int*   cnt    = (int*)  carve((size_t)n * 4);
  int*   rowp   = (int*)  carve((size_t)(n + 1) * 4);
  int*   cursor = (int*)  carve((size_t)n * 4);
  int*   col    = (int*)  carve((size_t)e * 4);
  float* wgt    = (float*)carve((size_t)e * 4);
  float* xA     = (float*)carve((size_t)n * DFEAT * 4);
  float* xB     = (float*)carve((size_t)n * DFEAT * 4);
  float* h      = (float*)carve((size_t)n * DFEAT * 4);

  const long long nh = (long long)n * DFEAT;

  ssgc_init<<<(int)((nh + 255) / 256), 256, 0, stream>>>(dinv, cnt, h, n, nh);
  ssgc_count<<<(e + 255) / 256, 256, 0, stream>>>(dst, dinv, cnt, e);
  ssgc_dinv<<<(n + 255) / 256, 256, 0, stream>>>(dinv, n);
  ssgc_scan<<<1, 1024, 0, stream>>>(cnt, rowp, cursor, n, e);
  ssgc_scatter<<<(e + 255) / 256, 256, 0, stream>>>(src, dst, dinv, cursor, col, wgt, e);

  // 16 propagation rounds, ping-pong x, fused h update
  const float* xin = feat;
  float* xout = xA;
  for (int k = 0; k < KSTEP; ++k) {
    int waves = n;                                   // one wave32 per node
    int blocks = (waves * 32 + 255) / 256;
    ssgc_step<<<blocks, 256, 0, stream>>>(xin, xout, h, feat, dinv, rowp, col, wgt, n);
    xin = xout;
    xout = (xout == xA) ? xB : xA;
  }

  // final fp32 WMMA GEMM
  int tiles = ((n + 15) / 16) * 3;
  int blocks = (tiles + 7) / 8;                      // 8 waves per 256-thread block
  ssgc_gemm<<<blocks, 256, 0, stream>>>(h, W, bias, out, n);
}